// GraphNeuralNetwork_23545010717206
// MI455X (gfx1250) — compile-verified
//
#include <hip/hip_runtime.h>
#include <hip/hip_bf16.h>

// GCN forward on MI455X (gfx1250, wave32).
// Bottleneck analysis: scatter-add of 1.7M edges x 128 f32 features per layer
// (~870MB gathers + ~218M f32 atomics per layer). Whole working set (~150MB)
// fits in the 192MB L2, so this runs at L2 speed. GEMMs are only 7.4 GFLOP
// total -> use the native f32 WMMA path (v_wmma_f32_16x16x4_f32) for full
// accuracy at negligible cost.

typedef float v2f __attribute__((ext_vector_type(2)));
typedef float v8f __attribute__((ext_vector_type(8)));

#define FOUT 128

// ---------------- degree / norm precompute ----------------

__global__ void init_deg_kernel(float* __restrict__ deg, int n) {
  int i = blockIdx.x * blockDim.x + threadIdx.x;
  if (i < n) deg[i] = 1.0f;  // self-loop contributes 1 to every node's degree
}

__global__ void deg_edges_kernel(const int* __restrict__ dst, float* __restrict__ deg, int E) {
  int e = blockIdx.x * blockDim.x + threadIdx.x;
  if (e < E) atomicAdd(&deg[dst[e]], 1.0f);
}

__global__ void dinv_kernel(float* __restrict__ deg, int n) {
  int i = blockIdx.x * blockDim.x + threadIdx.x;
  if (i < n) {
    float d = deg[i];
    deg[i] = (d > 0.0f) ? rsqrtf(d) : 0.0f;  // deg >= 1 always (self-loop)
  }
}

__global__ void norm_kernel(const int* __restrict__ src, const int* __restrict__ dst,
                            const float* __restrict__ dinv, float* __restrict__ norm, int E) {
  int e = blockIdx.x * blockDim.x + threadIdx.x;
  if (e < E) norm[e] = dinv[src[e]] * dinv[dst[e]];
}

// ---------------- dense GEMM: C[N,128] = A[N,K] x W[K,128] via f32 WMMA ----
// Block = 256 threads = 8 wave32; block covers a 16-row x 128-col C panel,
// each wave owns one 16x16 tile. K staged through LDS in chunks of 32 with
// padded strides (33 / 132) to avoid 64-bank LDS conflicts.
//
// WMMA f32 16x16x4 operand layout (ISA 7.12.2):
//   A (16x4, 2 VGPR):  lanes 0-15 -> M=lane, {K=k+0,k+1}; lanes 16-31 -> {K=k+2,k+3}
//   B (4x16, 2 VGPR):  lanes 0-15 -> N=lane, rows {k+0,k+1}; lanes 16-31 -> {k+2,k+3}
//   C/D (16x16, 8 VGPR): VGPR i -> M=i (lanes 0-15), M=i+8 (lanes 16-31), N=lane&15

template <int K>
__global__ void gemm_wmma_kernel(const float* __restrict__ A, const float* __restrict__ W,
                                 float* __restrict__ C) {
  __shared__ float ldsW[32 * 132];  // 32 K-rows of W, padded stride 132
  __shared__ float ldsA[16 * 33];   // 16 M-rows x 32 K, padded stride 33

  const int tid  = threadIdx.x;
  const int lane = tid & 31;
  const int wv   = tid >> 5;          // 0..7 -> which 16-wide N tile
  const int m0   = blockIdx.x * 16;
  const int n0   = wv * 16;

  v8f acc = {};

  const int row   = lane & 15;
  const int khalf = (lane >> 4) << 1;  // 0 for lanes 0-15, 2 for lanes 16-31

  for (int k0 = 0; k0 < K; k0 += 32) {
    for (int idx = tid; idx < 32 * FOUT; idx += 256) {
      int r = idx >> 7, c = idx & 127;
      ldsW[r * 132 + c] = W[(k0 + r) * FOUT + c];
    }
    for (int idx = tid; idx < 16 * 32; idx += 256) {
      int r = idx >> 5, c = idx & 31;
      ldsA[r * 33 + c] = A[(size_t)(m0 + r) * K + k0 + c];
    }
    __syncthreads();

#pragma unroll
    for (int kk = 0; kk < 32; kk += 4) {
      v2f a, b;
      a.x = ldsA[row * 33 + kk + khalf];
      a.y = ldsA[row * 33 + kk + khalf + 1];
      b.x = ldsW[(kk + khalf) * 132 + n0 + row];
      b.y = ldsW[(kk + khalf + 1) * 132 + n0 + row];
      acc = __builtin_amdgcn_wmma_f32_16x16x4_f32(false, a, false, b,
                                                  (short)0, acc, false, false);
    }
    __syncthreads();
  }

  const int col   = n0 + (lane & 15);
  const int rbase = m0 + ((lane >> 4) << 3);
#pragma unroll
  for (int i = 0; i < 8; ++i)
    C[(size_t)(rbase + i) * FOUT + col] = acc[i];
}

// ---------------- edge scatter: out[dst] += norm * h[src] ------------------
// One wave32 per edge; lane l handles features [4l, 4l+4). The 512B row
// gather is one coalesced float4/lane load (L2-resident), followed by 4
// global_atomic_add_f32 per lane.

__global__ void scatter_edges_kernel(const int* __restrict__ src, const int* __restrict__ dst,
                                     const float* __restrict__ norm,
                                     const float* __restrict__ h,
                                     float* __restrict__ out, int E) {
  int gw   = (int)((blockIdx.x * (size_t)blockDim.x + threadIdx.x) >> 5);
  int lane = threadIdx.x & 31;
  if (gw >= E) return;
  int s = src[gw];
  int d = dst[gw];
  float nv = norm[gw];
  const float4 hv = *(const float4*)(h + (size_t)s * FOUT + lane * 4);
  float* o = out + (size_t)d * FOUT + lane * 4;
  atomicAdd(o + 0, nv * hv.x);
  atomicAdd(o + 1, nv * hv.y);
  atomicAdd(o + 2, nv * hv.z);
  atomicAdd(o + 3, nv * hv.w);
}

// ---------------- self-loop + bias + ReLU (fused epilogue) -----------------

__global__ void finish_kernel(const float* __restrict__ h, const float* __restrict__ dinv,
                              const float* __restrict__ bias, float* __restrict__ out,
                              size_t total, int do_relu) {
  size_t idx = (size_t)blockIdx.x * blockDim.x + threadIdx.x;
  if (idx >= total) return;
  int i = (int)(idx >> 7);
  int f = (int)(idx & 127);
  float di = dinv[i];
  float v = out[idx] + di * di * h[idx] + bias[f];
  out[idx] = do_relu ? fmaxf(v, 0.0f) : v;
}

// ---------------- mean pool + FC: out = (1/N) sum_i <h[i], fcW> + fcb ------

__global__ void pool_stage1_kernel(const float* __restrict__ h, const float* __restrict__ fcW,
                                   float* __restrict__ partials, int n) {
  int lane  = threadIdx.x & 31;
  int wvIn  = threadIdx.x >> 5;
  int wpb   = blockDim.x >> 5;
  int gwave = blockIdx.x * wpb + wvIn;
  int nwave = gridDim.x * wpb;

  const float4 w4 = *(const float4*)(fcW + lane * 4);
  float acc = 0.0f;
  for (int i = gwave; i < n; i += nwave) {
    const float4 hv = *(const float4*)(h + (size_t)i * FOUT + lane * 4);
    acc += hv.x * w4.x + hv.y * w4.y + hv.z * w4.z + hv.w * w4.w;
  }
#pragma unroll
  for (int off = 16; off > 0; off >>= 1) acc += __shfl_xor(acc, off, 32);

  __shared__ float smem[8];
  if (lane == 0) smem[wvIn] = acc;
  __syncthreads();
  if (threadIdx.x == 0) {
    float s = 0.0f;
    for (int w = 0; w < wpb; ++w) s += smem[w];
    partials[blockIdx.x] = s;
  }
}

__global__ void pool_stage2_kernel(const float* __restrict__ partials,
                                   const float* __restrict__ fcb,
                                   float* __restrict__ out, int nPart, float invN) {
  __shared__ float smem[256];
  float v = (threadIdx.x < nPart) ? partials[threadIdx.x] : 0.0f;
  smem[threadIdx.x] = v;
  __syncthreads();
  for (int s = 128; s > 0; s >>= 1) {
    if ((int)threadIdx.x < s) smem[threadIdx.x] += smem[threadIdx.x + s];
    __syncthreads();
  }
  if (threadIdx.x == 0) out[0] = smem[0] * invN + fcb[0];
}

// ---------------- driver ----------------------------------------------------

extern "C" void kernel_launch(void* const* d_in, const int* in_sizes, int n_in,
                              void* d_out, int out_size, void* d_ws, size_t ws_size,
                              hipStream_t stream) {
  const float* x   = (const float*)d_in[0];
  const int*   ei  = (const int*)d_in[1];   // edge_index [2,E], int32 per harness mapping
  const float* W0  = (const float*)d_in[2];
  const float* b0  = (const float*)d_in[3];
  const float* W1  = (const float*)d_in[4];
  const float* b1  = (const float*)d_in[5];
  const float* W2  = (const float*)d_in[6];
  const float* b2  = (const float*)d_in[7];
  const float* fcW = (const float*)d_in[8];
  const float* fcb = (const float*)d_in[9];
  float* out = (float*)d_out;

  const int N = in_sizes[0] / 32;   // x is [N, 32]
  const int E = in_sizes[1] / 2;    // edge_index is [2, E]
  const int* src = ei;
  const int* dst = ei + E;

  // workspace layout (floats): dinv[N] | norm[E] | hA[N*128] | hB[N*128] | partials[256]
  float* ws       = (float*)d_ws;
  float* dinv     = ws;
  float* norm     = ws + N;
  float* hA       = norm + E;
  float* hB       = hA + (size_t)N * FOUT;
  float* partials = hB + (size_t)N * FOUT;

  const size_t featTotal = (size_t)N * FOUT;
  const size_t featBytes = featTotal * sizeof(float);

  const int nodeBlocks = (N + 255) / 256;
  const int edgeBlocks = (E + 255) / 256;
  const int gemmBlocks = N / 16;                      // N = 100000, multiple of 16
  const int scatBlocks = (E + 7) / 8;                 // 8 edges (waves) per 256-thread block
  const int finBlocks  = (int)((featTotal + 255) / 256);

  // degrees -> dinv -> per-edge norms (self-loops handled analytically later)
  init_deg_kernel<<<nodeBlocks, 256, 0, stream>>>(dinv, N);
  deg_edges_kernel<<<edgeBlocks, 256, 0, stream>>>(dst, dinv, E);
  dinv_kernel<<<nodeBlocks, 256, 0, stream>>>(dinv, N);
  norm_kernel<<<edgeBlocks, 256, 0, stream>>>(src, dst, dinv, norm, E);

  // ---- layer 0: 32 -> 128, ReLU ----
  gemm_wmma_kernel<32><<<gemmBlocks, 256, 0, stream>>>(x, W0, hA);
  hipMemsetAsync(hB, 0, featBytes, stream);
  scatter_edges_kernel<<<scatBlocks, 256, 0, stream>>>(src, dst, norm, hA, hB, E);
  finish_kernel<<<finBlocks, 256, 0, stream>>>(hA, dinv, b0, hB, featTotal, 1);

  // ---- layer 1: 128 -> 128, ReLU ----
  gemm_wmma_kernel<128><<<gemmBlocks, 256, 0, stream>>>(hB, W1, hA);
  hipMemsetAsync(hB, 0, featBytes, stream);   // hB already consumed by GEMM (stream order)
  scatter_edges_kernel<<<scatBlocks, 256, 0, stream>>>(src, dst, norm, hA, hB, E);
  finish_kernel<<<finBlocks, 256, 0, stream>>>(hA, dinv, b1, hB, featTotal, 1);

  // ---- layer 2: 128 -> 128, no activation ----
  gemm_wmma_kernel<128><<<gemmBlocks, 256, 0, stream>>>(hB, W2, hA);
  hipMemsetAsync(hB, 0, featBytes, stream);
  scatter_edges_kernel<<<scatBlocks, 256, 0, stream>>>(src, dst, norm, hA, hB, E);
  finish_kernel<<<finBlocks, 256, 0, stream>>>(hA, dinv, b2, hB, featTotal, 0);

  // ---- mean pool + FC ----
  pool_stage1_kernel<<<256, 256, 0, stream>>>(hB, fcW, partials, N);
  pool_stage2_kernel<<<1, 256, 0, stream>>>(partials, fcb, out, 256, 1.0f / (float)N);
}